// RTGB_78451872628994
// MI455X (gfx1250) — compile-verified
//
#include <hip/hip_runtime.h>
#include <math.h>

// ---------------------------------------------------------------------------
// Problem: B=8, C=64, W=256, H=256 (fp32).
// out[b,c,w,h] = sig(Wc@mean_wh + bc)[b,c] * sig(Ww@mean_ch + bw)[b,w]
//                                          * sig(Wh@mean_cw + bh)[b,h]
// Bandwidth-bound (134MB in, 134MB out -> ~11.5us floor @ 23.3TB/s; this
// pipeline moves ~400MB ~= 17us). GEMMs use V_WMMA_F32_16X16X4_F32 (fp32).
// ---------------------------------------------------------------------------

typedef __attribute__((ext_vector_type(2))) float v2f;
typedef __attribute__((ext_vector_type(8))) float v8f;

#define NB 8
#define NC 64
#define NW 256
#define NH 256

// ---------------- zero init (sumH is accumulated with atomics) --------------
__global__ void rtgb_zero(float* p, int n) {
    int i = blockIdx.x * blockDim.x + threadIdx.x;
    if (i < n) p[i] = 0.0f;
}

// ------------- pool over (W,H): one block per (b,c) plane, float4 loads -----
__global__ void rtgb_pool_c(const float* __restrict__ x, float* __restrict__ sumC) {
    const int c = blockIdx.x, b = blockIdx.y, t = threadIdx.x;
    const float4* p = reinterpret_cast<const float4*>(x + (size_t)(b * NC + c) * (NW * NH));
    float acc = 0.0f;
    // 65536 floats = 16384 float4, 256 threads -> 64 iters, fully coalesced
    for (int i = t; i < (NW * NH / 4); i += 256) {
        float4 v = p[i];
        acc += (v.x + v.y) + (v.z + v.w);
    }
    __shared__ float lds[256];
    lds[t] = acc;
    __syncthreads();
    for (int s = 128; s > 0; s >>= 1) {
        if (t < s) lds[t] += lds[t + s];
        __syncthreads();
    }
    if (t == 0) sumC[b * NC + c] = lds[0];   // raw sum; scaled in gate kernel
}

// --- pool over C: block per (b,w), thread = h. Coalesced (h contiguous). ----
// Block-reduce across h -> sumW[b,w]; per-thread atomic over w-blocks -> sumH[b,h].
__global__ void rtgb_pool_wh(const float* __restrict__ x,
                             float* __restrict__ sumW, float* __restrict__ sumH) {
    const int w = blockIdx.x, b = blockIdx.y, t = threadIdx.x;  // t = h
    const float* p = x + (size_t)b * NC * NW * NH + (size_t)w * NH + t;
    float acc = 0.0f;
    #pragma unroll 4
    for (int c = 0; c < NC; ++c) acc += p[(size_t)c * (NW * NH)];
    atomicAdd(&sumH[b * NH + t], acc);       // 256 colliding adds per slot, tiny
    __shared__ float lds[256];
    lds[t] = acc;
    __syncthreads();
    for (int s = 128; s > 0; s >>= 1) {
        if (t < s) lds[t] += lds[t + s];
        __syncthreads();
    }
    if (t == 0) sumW[b * NW + w] = lds[0];
}

// ---------------- gate: D = sigmoid(Wmat @ pooled^T + bias) -----------------
// One wave (32 threads) per 16-output tile. V_WMMA_F32_16X16X4_F32:
//   A (16x4, M x K):  lanes 0-15 hold M=lane, K={0,1}; lanes 16-31 hold K={2,3}
//   B (4x16, K x N):  lanes 0-15 hold N=lane, K={0,1}; lanes 16-31 hold K={2,3}
//   D (16x16):        VGPR r: M = r + 8*(lane>=16), N = lane&15
// N = batch index (8 valid, 8 zero-padded). Padding is branch-free: padded
// lanes read a clamped-but-valid row and multiply by lscale = 0.
__global__ void rtgb_gate(const float* __restrict__ Wmat, const float* __restrict__ bias,
                          const float* __restrict__ pooledRaw, float scale,
                          float* __restrict__ gateOut, int O) {
    const int lane = threadIdx.x;        // wave32
    const int half = lane >> 4;          // 0: K {0,1}   1: K {2,3}
    const int lr   = lane & 15;          // A: M row     B: N col (= batch)
    const int o0   = blockIdx.x * 16;
    const int K    = O;                  // square weight matrices

    const float lscale = (lr < NB) ? scale : 0.0f;           // zero pad lanes
    const float* arow = Wmat + (size_t)(o0 + lr) * K + half * 2;
    const float* brow = pooledRaw + (size_t)(lr & (NB - 1)) * K + half * 2;

    v8f acc = {};
    for (int k0 = 0; k0 < K; k0 += 4) {
        const v2f a    = *reinterpret_cast<const v2f*>(arow + k0);  // b64 load
        const v2f braw = *reinterpret_cast<const v2f*>(brow + k0);  // b64 load
        v2f bm;
        bm.x = braw.x * lscale;          // B[k][n] = mean[n][k], 0 for n>=8
        bm.y = braw.y * lscale;
        acc = __builtin_amdgcn_wmma_f32_16x16x4_f32(
            /*neg_a=*/false, a, /*neg_b=*/false, bm,
            /*c_mod=*/(short)0, acc, /*reuse_a=*/false, /*reuse_b=*/false);
    }

    // writeback: element (m, n) with m = r + 8*half, n = lr (batch)
    if (lr < NB) {
        #pragma unroll
        for (int r = 0; r < 8; ++r) {
            const int m = r + half * 8;
            const float v = acc[r] + bias[o0 + m];
            gateOut[lr * O + o0 + m] = 1.0f / (1.0f + __expf(-v));
        }
    }
}

// ---------------- outer product: out[b,c,w,h] = gc*gw*gh, float4 stores -----
__global__ void rtgb_outer(const float* __restrict__ gc, const float* __restrict__ gw,
                           const float* __restrict__ gh, float4* __restrict__ out) {
    const int idx = blockIdx.x * blockDim.x + threadIdx.x;  // float4 index
    const int e = idx << 2;                                  // element index
    const int h = e & (NH - 1);
    const int w = (e >> 8) & (NW - 1);
    const int c = (e >> 16) & (NC - 1);
    const int b = e >> 22;                                   // 64*256*256 = 2^22
    const float s = gc[b * NC + c] * gw[b * NW + w];
    const float4 g4 = reinterpret_cast<const float4*>(gh + b * NH)[h >> 2];
    out[idx] = make_float4(s * g4.x, s * g4.y, s * g4.z, s * g4.w);
}

// ---------------------------------------------------------------------------
extern "C" void kernel_launch(void* const* d_in, const int* in_sizes, int n_in,
                              void* d_out, int out_size, void* d_ws, size_t ws_size,
                              hipStream_t stream) {
    (void)in_sizes; (void)n_in; (void)out_size; (void)ws_size;
    const float* x  = (const float*)d_in[0];
    const float* Wc = (const float*)d_in[1];
    const float* bc = (const float*)d_in[2];
    const float* Ww = (const float*)d_in[3];
    const float* bw = (const float*)d_in[4];
    const float* Wh = (const float*)d_in[5];
    const float* bh = (const float*)d_in[6];
    float* out = (float*)d_out;
    float* ws  = (float*)d_ws;

    float* sumC = ws;            // 512
    float* sumW = ws + 512;      // 2048
    float* sumH = ws + 2560;     // 2048
    float* gC   = ws + 4608;     // 512
    float* gW   = ws + 5120;     // 2048
    float* gH   = ws + 7168;     // 2048   (total 36 KB of workspace)

    rtgb_zero<<<NB, 256, 0, stream>>>(sumH, NB * NH);
    rtgb_pool_c<<<dim3(NC, NB), 256, 0, stream>>>(x, sumC);
    rtgb_pool_wh<<<dim3(NW, NB), 256, 0, stream>>>(x, sumW, sumH);

    // mean scales: /(W*H)=1/65536 for C-gate, /(C*H)=/(C*W)=1/16384 for W/H-gates
    rtgb_gate<<<NC / 16, 32, 0, stream>>>(Wc, bc, sumC, 1.0f / 65536.0f, gC, NC);
    rtgb_gate<<<NW / 16, 32, 0, stream>>>(Ww, bw, sumW, 1.0f / 16384.0f, gW, NW);
    rtgb_gate<<<NH / 16, 32, 0, stream>>>(Wh, bh, sumH, 1.0f / 16384.0f, gH, NH);

    // 33.5M elements / 4 per thread / 256 per block = 32768 blocks
    rtgb_outer<<<(NB * NC * NW * NH) / (4 * 256), 256, 0, stream>>>(gC, gW, gH, (float4*)out);
}